// DecoderAttn_66821101191347
// MI455X (gfx1250) — compile-verified
//
#include <hip/hip_runtime.h>
#include <hip/hip_bf16.h>

typedef __attribute__((ext_vector_type(2))) float v2f;
typedef __attribute__((ext_vector_type(8))) float v8f;

#define B_ 32
#define T_ 2048
#define H_ 1024
#define SLICES 16  // energy blocks per batch

// ---------------------------------------------------------------------------
// Kernel 1: v = dec @ W   (v[b,h] = sum_o dec[b,o] * W[o*H + h])
// M=32, K=1024, N=1024 via V_WMMA_F32_16X16X4_F32.
// 128 waves total; each wave owns one 16x16 tile (btile in {0,1}, ntile in 0..63).
// A (16x4 f32, 2 VGPRs): lanes 0-15 -> M=lane, K=k0+{0,1}; lanes 16-31 -> K=k0+{2,3}.
// B (4x16 f32, 2 VGPRs): VGPR r -> K = k0 + r + 2*(lane>=16), N = lane&15.
// D (16x16 f32, 8 VGPRs): VGPR r -> M = r + 8*(lane>=16), N = lane&15.
// ---------------------------------------------------------------------------
__global__ __launch_bounds__(256) void DecoderAttn_proj_wmma(
    const float* __restrict__ dec, const float* __restrict__ W,
    float* __restrict__ v)
{
    const int lane  = threadIdx.x & 31;
    const int wid   = blockIdx.x * (blockDim.x >> 5) + (threadIdx.x >> 5);
    const int btile = wid & 1;
    const int ntile = wid >> 1;
    const int hlf   = lane >> 4;   // 0 or 1
    const int l15   = lane & 15;

    const int arow = btile * 16 + l15;   // dec row (M)
    const int ncol = ntile * 16 + l15;   // v/W column (N)

    v8f acc = {};
#pragma unroll 4
    for (int k0 = 0; k0 < H_; k0 += 4) {
        const int ka = k0 + hlf * 2;                       // even -> 8B aligned
        v2f a = *(const v2f*)(dec + arow * H_ + ka);       // A[M=arow, K=ka..ka+1]
        v2f bm;
        bm.x = W[(ka    ) * H_ + ncol];                    // B[K=ka,   N=ncol]
        bm.y = W[(ka + 1) * H_ + ncol];                    // B[K=ka+1, N=ncol]
        acc = __builtin_amdgcn_wmma_f32_16x16x4_f32(
            /*neg_a=*/false, a, /*neg_b=*/false, bm,
            /*c_mod=*/(short)0, acc, /*reuse_a=*/false, /*reuse_b=*/false);
    }
#pragma unroll
    for (int r = 0; r < 8; ++r) {
        const int m = btile * 16 + r + 8 * hlf;
        v[m * H_ + ncol] = acc[r];
    }
}

// ---------------------------------------------------------------------------
// Kernel 2: c[b] = dec[b] . bias   (one wave per batch)
// ---------------------------------------------------------------------------
__global__ __launch_bounds__(32) void DecoderAttn_cdot(
    const float* __restrict__ dec, const float* __restrict__ bias,
    float* __restrict__ c)
{
    const int b = blockIdx.x;
    const int lane = threadIdx.x;
    float s = 0.0f;
#pragma unroll
    for (int j = lane; j < H_; j += 32) s += dec[b * H_ + j] * bias[j];
#pragma unroll
    for (int off = 16; off > 0; off >>= 1) s += __shfl_xor(s, off, 32);
    if (lane == 0) c[b] = s;
}

// ---------------------------------------------------------------------------
// Kernel 3: energies[b,t] = enc[b,t] . v[b] + c[b]   (bandwidth kernel, 256 MB)
// grid = (SLICES, B); block = 256 (8 waves); wave-per-t, lane-strided float4
// loads so each wave issues contiguous 512B global_load_b128 bursts.
// v[b] cached in 8 float4 registers per lane, amortized over 16 t's per wave.
// ---------------------------------------------------------------------------
__global__ __launch_bounds__(256) void DecoderAttn_energy(
    const float* __restrict__ enc, const float* __restrict__ v,
    const float* __restrict__ c, float* __restrict__ energies)
{
    const int b    = blockIdx.y;
    const int lane = threadIdx.x & 31;
    const int wave = threadIdx.x >> 5;
    const int tPerBlock = T_ / SLICES;          // 128
    const int tPerWave  = tPerBlock / 8;        // 16
    const int tBase = blockIdx.x * tPerBlock + wave * tPerWave;

    const float4* v4 = (const float4*)(v + b * H_);
    float4 vv[8];
#pragma unroll
    for (int j = 0; j < 8; ++j) vv[j] = v4[lane + 32 * j];
    const float cb = c[b];

    for (int ti = 0; ti < tPerWave; ++ti) {
        const int t = tBase + ti;
        const float4* e4 = (const float4*)(enc + ((size_t)b * T_ + t) * H_);
        float acc = 0.0f;
#pragma unroll
        for (int j = 0; j < 8; ++j) {
            const float4 e = e4[lane + 32 * j];
            acc += e.x * vv[j].x + e.y * vv[j].y + e.z * vv[j].z + e.w * vv[j].w;
        }
#pragma unroll
        for (int off = 16; off > 0; off >>= 1) acc += __shfl_xor(acc, off, 32);
        if (lane == 0) energies[b * T_ + t] = acc + cb;
    }
}

// ---------------------------------------------------------------------------
// Kernel 4: softmax over T per batch. One 256-thread block per batch,
// 8 values per thread held in registers, LDS tree reductions.
// ---------------------------------------------------------------------------
__global__ __launch_bounds__(256) void DecoderAttn_softmax(
    const float* __restrict__ energies, float* __restrict__ out)
{
    const int b   = blockIdx.x;
    const int tid = threadIdx.x;
    __shared__ float red[256];

    float e[8];
    float m = -__builtin_inff();
#pragma unroll
    for (int j = 0; j < 8; ++j) {
        e[j] = energies[b * T_ + tid + 256 * j];
        m = fmaxf(m, e[j]);
    }
    red[tid] = m;
    __syncthreads();
    for (int s = 128; s > 0; s >>= 1) {
        if (tid < s) red[tid] = fmaxf(red[tid], red[tid + s]);
        __syncthreads();
    }
    m = red[0];
    __syncthreads();

    float sum = 0.0f;
#pragma unroll
    for (int j = 0; j < 8; ++j) {
        e[j] = __expf(e[j] - m);
        sum += e[j];
    }
    red[tid] = sum;
    __syncthreads();
    for (int s = 128; s > 0; s >>= 1) {
        if (tid < s) red[tid] += red[tid + s];
        __syncthreads();
    }
    const float inv = 1.0f / red[0];
#pragma unroll
    for (int j = 0; j < 8; ++j) out[b * T_ + tid + 256 * j] = e[j] * inv;
}

extern "C" void kernel_launch(void* const* d_in, const int* in_sizes, int n_in,
                              void* d_out, int out_size, void* d_ws, size_t ws_size,
                              hipStream_t stream)
{
    const float* dec  = (const float*)d_in[0];  // [B,H]
    const float* enc  = (const float*)d_in[1];  // [B,T,H]
    const float* W    = (const float*)d_in[2];  // [H,H]
    const float* bias = (const float*)d_in[3];  // [H]
    float* out = (float*)d_out;                 // [B,T,1] f32

    float* v        = (float*)d_ws;             // B*H
    float* c        = v + B_ * H_;              // B
    float* energies = c + B_;                   // B*T

    // 1) v = dec @ W  (WMMA f32 16x16x4) : 128 waves = 16 blocks x 8 waves
    DecoderAttn_proj_wmma<<<16, 256, 0, stream>>>(dec, W, v);
    // 2) c[b] = dec[b] . bias
    DecoderAttn_cdot<<<B_, 32, 0, stream>>>(dec, bias, c);
    // 3) energies[b,t] = enc[b,t].v[b] + c[b]  (streams the 256MB tensor once)
    DecoderAttn_energy<<<dim3(SLICES, B_), 256, 0, stream>>>(enc, v, c, energies);
    // 4) softmax over T per batch -> d_out
    DecoderAttn_softmax<<<B_, 256, 0, stream>>>(energies, out);
}